// RegionProposalNetwork_43473658970682
// MI455X (gfx1250) — compile-verified
//
#include <hip/hip_runtime.h>
#include <stdint.h>

// ---------------- problem constants (from reference) ----------------
#define N_IMG     2
#define R_TOT     159882
#define NUM_LVL   5
#define PRE_NMS   1000
#define POST_NMS  1000
#define K_TOT     4507          // 1000+1000+1000+1000+507
#define SORT_N    8192          // next pow2 >= K_TOT
#define IMG_WH    800.0f
#define NMS_TH    0.7f
#define MIN_SIZE  1e-3f
#define BBOX_CLIP 4.135166556742356f   // ln(1000/16)

__constant__ int c_lvl_off[NUM_LVL + 1]  = {0, 120000, 150000, 157500, 159375, 159882};
__constant__ int c_slot_off[NUM_LVL + 1] = {0, 1000, 2000, 3000, 4000, 4507};

// -------- CDNA5 async global->LDS path (guarded; falls back cleanly) --------
#if defined(__has_builtin)
#if __has_builtin(__builtin_amdgcn_global_load_async_to_lds_b128) && \
    __has_builtin(__builtin_amdgcn_s_wait_asynccnt)
#define HAVE_ASYNC_LDS 1
#endif
#endif
#ifndef HAVE_ASYNC_LDS
#define HAVE_ASYNC_LDS 0
#endif

// Builtin parameter type on this toolchain (from round-1 diagnostic):
//   'int __attribute__((__vector_size__(4 * sizeof(int)))) *'
typedef int v4i_async __attribute__((vector_size(16)));

// monotonic float -> uint key (larger float => larger key)
__device__ __forceinline__ unsigned key_of(float f) {
  unsigned u = __float_as_uint(f);
  return (u & 0x80000000u) ? ~u : (u | 0x80000000u);
}

// ====================================================================
// Kernel 1: exact per-(image,level) top-k via 8-bit radix select + compaction
// grid = N_IMG*NUM_LVL blocks, 1024 threads
// ====================================================================
__global__ __launch_bounds__(1024) void rpn_topk(const float* __restrict__ obj,
                                                 unsigned* __restrict__ ckey,
                                                 int* __restrict__ cidx) {
  const int img = blockIdx.x / NUM_LVL;
  const int lvl = blockIdx.x % NUM_LVL;
  const int off = c_lvl_off[lvl];
  const int n   = c_lvl_off[lvl + 1] - off;
  const int k   = c_slot_off[lvl + 1] - c_slot_off[lvl];
  const float* __restrict__ o = obj + (size_t)img * R_TOT + off;
  const int t = threadIdx.x;

  __shared__ int hist[256];
  __shared__ int s_bucket, s_acc;

  unsigned prefix = 0u;
  int need = k;                 // entries still needed at/below current prefix
  for (int shift = 24; shift >= 0; shift -= 8) {
    if (t < 256) hist[t] = 0;
    __syncthreads();
    const unsigned hmask = (shift == 24) ? 0u : (0xFFFFFFFFu << (shift + 8));
    for (int i = t; i < n; i += 1024) {
      if (i + 8192 < n) __builtin_prefetch(o + i + 8192, 0, 1);
      unsigned key = key_of(o[i]);
      if ((key & hmask) == prefix) atomicAdd(&hist[(key >> shift) & 255], 1);
    }
    __syncthreads();
    if (t == 0) {
      int acc = 0, b = 255;
      for (; b > 0; --b) {
        int h = hist[b];
        if (acc + h >= need) break;
        acc += h;
      }
      s_bucket = b;
      s_acc = acc;
    }
    __syncthreads();
    prefix |= ((unsigned)s_bucket) << shift;
    need -= s_acc;
    __syncthreads();
  }
  const unsigned T = prefix;          // threshold key
  const int cnt_gt = k - need;        // # strictly greater than T

  __shared__ int sgt, seq;
  if (t == 0) { sgt = 0; seq = 0; }
  __syncthreads();

  unsigned* __restrict__ okey = ckey + (size_t)img * K_TOT + c_slot_off[lvl];
  int* __restrict__ oidx      = cidx + (size_t)img * K_TOT + c_slot_off[lvl];
  for (int i = t; i < n; i += 1024) {
    unsigned key = key_of(o[i]);
    int pos = -1;
    if (key > T) {
      pos = atomicAdd(&sgt, 1);
    } else if (key == T) {
      int e = atomicAdd(&seq, 1);
      if (e < need) pos = cnt_gt + e;
    }
    if (pos >= 0) { okey[pos] = key; oidx[pos] = off + i; }
  }
}

// ====================================================================
// Kernel 2: per-image bitonic sort (descending by key) of K_TOT candidates
// grid = N_IMG blocks, 1024 threads, 64KB LDS
// ====================================================================
__global__ __launch_bounds__(1024) void rpn_sort(const unsigned* __restrict__ ckey,
                                                 const int* __restrict__ cidx,
                                                 int* __restrict__ sorted_r) {
  __shared__ unsigned skey[SORT_N];
  __shared__ int sidx[SORT_N];
  const int img = blockIdx.x;
  const int t = threadIdx.x;

  for (int i = t; i < SORT_N; i += 1024) {
    if (i < K_TOT) {
      skey[i] = ckey[(size_t)img * K_TOT + i];
      sidx[i] = cidx[(size_t)img * K_TOT + i];
    } else {
      skey[i] = 0u;     // sorts last (descending)
      sidx[i] = 0;
    }
  }
  __syncthreads();

  for (int size = 2; size <= SORT_N; size <<= 1) {
    for (int stride = size >> 1; stride > 0; stride >>= 1) {
      for (int j = t; j < SORT_N; j += 1024) {
        int x = j ^ stride;
        if (x > j) {
          bool descBlk = ((j & size) == 0);
          unsigned kj = skey[j], kx = skey[x];
          bool sw = descBlk ? (kj < kx) : (kj > kx);
          if (sw) {
            skey[j] = kx; skey[x] = kj;
            int tmp = sidx[j]; sidx[j] = sidx[x]; sidx[x] = tmp;
          }
        }
      }
      __syncthreads();
    }
  }
  for (int i = t; i < K_TOT; i += 1024) sorted_r[(size_t)img * K_TOT + i] = sidx[i];
}

// ====================================================================
// Kernel 3: decode + clip + sigmoid + validity for the sorted candidates
// ====================================================================
__global__ void rpn_decode(const float* __restrict__ obj,
                           const float* __restrict__ deltas,
                           const float* __restrict__ anchors,
                           const int* __restrict__ sorted_r,
                           float4* __restrict__ off_boxes,
                           float4* __restrict__ raw_boxes,
                           float* __restrict__ mscore) {
  const int tid = blockIdx.x * blockDim.x + threadIdx.x;
  if (tid >= N_IMG * K_TOT) return;
  const int img = tid / K_TOT;
  const int r = sorted_r[tid];

  const float a0 = anchors[4 * r + 0], a1 = anchors[4 * r + 1];
  const float a2 = anchors[4 * r + 2], a3 = anchors[4 * r + 3];
  const size_t db = ((size_t)img * R_TOT + r) * 4;
  const float dx = deltas[db + 0], dy = deltas[db + 1];
  float dw = deltas[db + 2], dh = deltas[db + 3];
  dw = fminf(dw, BBOX_CLIP);
  dh = fminf(dh, BBOX_CLIP);

  const float wa = a2 - a0, ha = a3 - a1;
  const float cxa = a0 + 0.5f * wa, cya = a1 + 0.5f * ha;
  const float cx = dx * wa + cxa, cy = dy * ha + cya;
  const float w = __expf(dw) * wa, h = __expf(dh) * ha;

  float x1 = fminf(fmaxf(cx - 0.5f * w, 0.0f), IMG_WH);
  float y1 = fminf(fmaxf(cy - 0.5f * h, 0.0f), IMG_WH);
  float x2 = fminf(fmaxf(cx + 0.5f * w, 0.0f), IMG_WH);
  float y2 = fminf(fmaxf(cy + 0.5f * h, 0.0f), IMG_WH);

  const int lvl = (r >= 159375) ? 4 : (r >= 157500) ? 3 : (r >= 150000) ? 2
                : (r >= 120000) ? 1 : 0;
  const float ofv = (float)lvl * (IMG_WH + 1.0f);   // batched-nms offset trick

  off_boxes[tid] = make_float4(x1 + ofv, y1 + ofv, x2 + ofv, y2 + ofv);
  raw_boxes[tid] = make_float4(x1, y1, x2, y2);

  const float ov = obj[(size_t)img * R_TOT + r];
  const float sc = 1.0f / (1.0f + __expf(-ov));
  const bool valid = (x2 - x1 >= MIN_SIZE) && (y2 - y1 >= MIN_SIZE) && (sc >= 0.0f);
  mscore[tid] = valid ? sc : -2.0f;   // sign encodes validity
}

// ====================================================================
// Kernel 4: greedy NMS, one workgroup per image. Accepted set lives in LDS;
// candidate chunks staged global->LDS via CDNA5 async loads when available.
// ====================================================================
__global__ __launch_bounds__(1024) void rpn_nms(const float4* __restrict__ off_boxes,
                                                const float4* __restrict__ raw_boxes,
                                                const float* __restrict__ mscore,
                                                float* __restrict__ out_boxes,
                                                float* __restrict__ out_scores) {
  const int img = blockIdx.x;
  const int t = threadIdx.x;

  __shared__ float ax1[POST_NMS], ay1[POST_NMS], ax2[POST_NMS], ay2[POST_NMS];
  __shared__ float4 stage[1024];
  __shared__ float smeta[1024];
  __shared__ int s_cnt;

  float* __restrict__ ob = out_boxes + (size_t)img * POST_NMS * 4;
  float* __restrict__ os = out_scores + (size_t)img * POST_NMS;
  for (int i = t; i < POST_NMS; i += 1024) {
    ob[4 * i + 0] = 0.0f; ob[4 * i + 1] = 0.0f;
    ob[4 * i + 2] = 0.0f; ob[4 * i + 3] = 0.0f;
    os[i] = -1.0f;
  }
  if (t == 0) s_cnt = 0;
  __syncthreads();

  const float4* __restrict__ cob = off_boxes + (size_t)img * K_TOT;
  const float* __restrict__ cms  = mscore + (size_t)img * K_TOT;

  for (int cb = 0; cb < K_TOT; cb += 1024) {
    const int cnum = min(1024, K_TOT - cb);
    if (t < cnum) {
#if HAVE_ASYNC_LDS
      __builtin_amdgcn_global_load_async_to_lds_b128(
          (v4i_async*)(cob + cb + t), (v4i_async*)(&stage[t]), 0, 0);
#else
      stage[t] = cob[cb + t];
#endif
      smeta[t] = cms[cb + t];
    }
#if HAVE_ASYNC_LDS
    __builtin_amdgcn_s_wait_asynccnt(0);
#endif
    __syncthreads();

    for (int i = 0; i < cnum; ++i) {
      const int cnt = s_cnt;             // uniform: guarded by trailing barrier
      if (cnt >= POST_NMS) break;        // further keeps can't reach output
      const float4 b = stage[i];
      const float areaB = (b.z - b.x) * (b.w - b.y);
      int flag = 0;
      for (int j = t; j < cnt; j += 1024) {
        const float ix1 = fmaxf(ax1[j], b.x);
        const float iy1 = fmaxf(ay1[j], b.y);
        const float ix2 = fminf(ax2[j], b.z);
        const float iy2 = fminf(ay2[j], b.w);
        const float iw = fmaxf(ix2 - ix1, 0.0f);
        const float ih = fmaxf(iy2 - iy1, 0.0f);
        const float inter = iw * ih;
        const float areaA = (ax2[j] - ax1[j]) * (ay2[j] - ay1[j]);
        const float iou = inter / (areaA + areaB - inter + 1e-9f);
        flag |= (iou > NMS_TH);
      }
      const int supp = __syncthreads_or(flag);
      if (t == 0 && !supp && smeta[i] >= 0.0f) {
        ax1[cnt] = b.x; ay1[cnt] = b.y; ax2[cnt] = b.z; ay2[cnt] = b.w;
        const float4 rb = raw_boxes[(size_t)img * K_TOT + cb + i];
        ob[4 * cnt + 0] = rb.x; ob[4 * cnt + 1] = rb.y;
        ob[4 * cnt + 2] = rb.z; ob[4 * cnt + 3] = rb.w;
        os[cnt] = smeta[i];
        s_cnt = cnt + 1;
      }
      __syncthreads();
    }
  }
}

// ====================================================================
// launch
// ====================================================================
static inline size_t align256(size_t x) { return (x + 255) & ~(size_t)255; }

extern "C" void kernel_launch(void* const* d_in, const int* in_sizes, int n_in,
                              void* d_out, int out_size, void* d_ws, size_t ws_size,
                              hipStream_t stream) {
  const float* obj     = (const float*)d_in[0];   // [2, R]
  const float* deltas  = (const float*)d_in[1];   // [2, R, 4]
  const float* anchors = (const float*)d_in[2];   // [R, 4]

  char* ws = (char*)d_ws;
  size_t off = 0;
  float4* off_boxes = (float4*)(ws + off); off += align256((size_t)N_IMG * K_TOT * sizeof(float4));
  float4* raw_boxes = (float4*)(ws + off); off += align256((size_t)N_IMG * K_TOT * sizeof(float4));
  unsigned* ckey    = (unsigned*)(ws + off); off += align256((size_t)N_IMG * K_TOT * sizeof(unsigned));
  int* cidx         = (int*)(ws + off);      off += align256((size_t)N_IMG * K_TOT * sizeof(int));
  int* sorted_r     = (int*)(ws + off);      off += align256((size_t)N_IMG * K_TOT * sizeof(int));
  float* mscore     = (float*)(ws + off);    off += align256((size_t)N_IMG * K_TOT * sizeof(float));
  (void)ws_size; (void)in_sizes; (void)n_in; (void)out_size;

  rpn_topk<<<N_IMG * NUM_LVL, 1024, 0, stream>>>(obj, ckey, cidx);
  rpn_sort<<<N_IMG, 1024, 0, stream>>>(ckey, cidx, sorted_r);
  const int total = N_IMG * K_TOT;
  rpn_decode<<<(total + 255) / 256, 256, 0, stream>>>(obj, deltas, anchors, sorted_r,
                                                      off_boxes, raw_boxes, mscore);
  float* out_b = (float*)d_out;                        // [2,1000,4]
  float* out_s = out_b + (size_t)N_IMG * POST_NMS * 4; // [2,1000]
  rpn_nms<<<N_IMG, 1024, 0, stream>>>(off_boxes, raw_boxes, mscore, out_b, out_s);
}